// ScaleShiftInvariantLoss_7069516169453
// MI455X (gfx1250) — compile-verified
//
#include <hip/hip_runtime.h>
#include <stdint.h>

#define B_ROWS 32
#define N_COLS (512 * 1024)
#define BPR 16                       // blocks per row (streaming kernels)
#define NBLK (B_ROWS * BPR)          // 512
#define TPB 256
#define TILE 4096                    // floats per staged tile (16 KB)
#define EPB (N_COLS / BPR)           // 32768 elements per block
#define TILES_PER_BLK (EPB / TILE)   // 8

// ---- workspace layout (dword offsets) ----
#define OFF_HISTP 0
#define OFF_HISTG (OFF_HISTP + B_ROWS * 2048)
#define OFF_CNT   (OFF_HISTG + B_ROWS * 2048)
#define OFF_GMINK (OFF_CNT   + B_ROWS)
#define OFF_GMAXK (OFF_GMINK + B_ROWS)
#define OFF_PREFP (OFF_GMAXK + B_ROWS)
#define OFF_RANKP (OFF_PREFP + B_ROWS)
#define OFF_PREFG (OFF_RANKP + B_ROWS)
#define OFF_RANKG (OFF_PREFG + B_ROWS)
#define OFF_TP    (OFF_RANKG + B_ROWS)
#define OFF_TG    (OFF_TP    + B_ROWS)
#define OFF_PSP   (OFF_TG    + B_ROWS)
#define OFF_PSG   (OFF_PSP   + NBLK)
#define OFF_PERR  (OFF_PSG   + NBLK)
#define WS_DWORDS (OFF_PERR  + NBLK)   // ~132896 dwords = ~520 KB

typedef __attribute__((ext_vector_type(4))) uint32_t u32x4;
typedef __attribute__((ext_vector_type(8))) uint32_t u32x8;
typedef __attribute__((ext_vector_type(2))) float v2f;
typedef __attribute__((ext_vector_type(8))) float v8f;

// ---- monotone float <-> uint key (ascending) ----
__device__ __forceinline__ uint32_t fkey(float f) {
  uint32_t u = __float_as_uint(f);
  return (u & 0x80000000u) ? ~u : (u | 0x80000000u);
}
__device__ __forceinline__ float funkey(uint32_t k) {
  uint32_t u = (k & 0x80000000u) ? (k & 0x7FFFFFFFu) : ~k;
  return __uint_as_float(u);
}

// Low 32 bits of a flat LDS-aperture address are the LDS byte offset.
__device__ __forceinline__ uint32_t lds_off(const void* p) {
  return (uint32_t)(uintptr_t)p;
}

// ---- Tensor Data Mover: 1-D tile of fp32, global -> LDS (async, TENSORcnt) ----
__device__ __forceinline__ void tdm_load_f32(uint32_t ldsAddr, const void* gaddr,
                                             uint32_t nElems) {
  uint64_t ga = (uint64_t)(uintptr_t)gaddr;
  u32x4 g0;
  g0[0] = 1u;                                             // count=1 valid descriptor
  g0[1] = ldsAddr;                                        // lds_addr
  g0[2] = (uint32_t)(ga & 0xFFFFFFFFu);                   // global_addr[31:0]
  g0[3] = (uint32_t)((ga >> 32) & 0x01FFFFFFu) | (2u << 30); // addr[56:32] | type=2
  u32x8 g1;
  g1[0] = 2u << 16;                                       // data_size = 4 bytes
  g1[1] = (nElems & 0xFFFFu) << 16;                       // tensor_dim0[15:0]
  g1[2] = ((nElems >> 16) & 0xFFFFu) | (1u << 16);        // tensor_dim0[31:16], tensor_dim1=1
  g1[3] = (nElems & 0xFFFFu) << 16;                       // tile_dim0
  g1[4] = 0;                                              // tile_dim1/2 unused
  g1[5] = nElems;                                         // tensor_dim0_stride lo
  g1[6] = 0;
  g1[7] = 0;
  asm volatile("tensor_load_to_lds %0, %1" :: "s"(g0), "s"(g1) : "memory");
}

__device__ __forceinline__ void tdm_issue_pair(float* sP, float* sG,
                                               const float* gP, const float* gG) {
  tdm_load_f32(lds_off(sP), gP, TILE);
  tdm_load_f32(lds_off(sG), gG, TILE);
}

// Double-buffered TDM pipeline: wave 0 keeps up to 2 tile-pairs in flight.
// TDM ops complete in order per wave, so waiting TENSORcnt<=2 guarantees the
// OLDER pair has fully landed in LDS while the newer pair still streams.
#define STREAM_TILES(...)                                                     \
  if (threadIdx.x < 32) tdm_issue_pair(sPbuf[0], sGbuf[0], gPb, gGb);         \
  for (int t = 0; t < TILES_PER_BLK; ++t) {                                   \
    const int cur = t & 1;                                                    \
    if (threadIdx.x < 32) {                                                   \
      if (t + 1 < TILES_PER_BLK) {                                            \
        tdm_issue_pair(sPbuf[cur ^ 1], sGbuf[cur ^ 1],                        \
                       gPb + (t + 1) * TILE, gGb + (t + 1) * TILE);           \
        __builtin_amdgcn_s_wait_tensorcnt(2);                                 \
      } else {                                                                \
        __builtin_amdgcn_s_wait_tensorcnt(0);                                 \
      }                                                                       \
    }                                                                         \
    __syncthreads();                                                          \
    const float* sP = sPbuf[cur];                                             \
    const float* sG = sGbuf[cur];                                             \
    __VA_ARGS__                                                               \
    __syncthreads();                                                          \
  }

// ---- workgroup sum reduction via V_WMMA_F32_16X16X4_F32; result on thread 0 ----
__device__ __forceinline__ float block_reduce_sum(float v, float* sRed) {
  sRed[threadIdx.x] = v;
  __syncthreads();
  float out = 0.0f;
#if __has_builtin(__builtin_amdgcn_wmma_f32_16x16x4_f32)
  if (threadIdx.x < 32) {            // wave 0, EXEC all ones
    v8f acc = {};
    v2f ones; ones.x = 1.0f; ones.y = 1.0f;
#pragma unroll
    for (int it = 0; it < 4; ++it) { // A = 64 partials, B = ones, C accumulates
      v2f a;
      a.x = sRed[it * 64 + threadIdx.x];
      a.y = sRed[it * 64 + 32 + threadIdx.x];
      acc = __builtin_amdgcn_wmma_f32_16x16x4_f32(false, a, false, ones,
                                                  (short)0, acc, false, false);
    }
    float part = acc[0] + acc[1] + acc[2] + acc[3] +
                 acc[4] + acc[5] + acc[6] + acc[7];   // D[m][0], m=0..7 (or 8..15)
    if (threadIdx.x == 0)  sRed[0] = part;
    if (threadIdx.x == 16) sRed[1] = part;            // same-wave DS ops are in order
    if (threadIdx.x == 0)  out = sRed[0] + sRed[1];
  }
#else
  for (int s = TPB / 2; s > 0; s >>= 1) {
    if (threadIdx.x < (unsigned)s) sRed[threadIdx.x] += sRed[threadIdx.x + s];
    __syncthreads();
  }
  out = sRed[0];
#endif
  __syncthreads();
  return out;
}

// ================= kernels =================

__global__ void k_init(uint32_t* ws) {
  int i = blockIdx.x * blockDim.x + threadIdx.x;
  if (i < WS_DWORDS) {
    uint32_t v = 0;
    if (i >= OFF_GMINK && i < OFF_GMINK + B_ROWS) v = 0xFFFFFFFFu;
    ws[i] = v;
  }
}

// Pass 1: count, gt min/max keys, level-0 (bits 31:21) histograms of pred & gt keys.
__global__ void __launch_bounds__(TPB) k_hist0(const float* __restrict__ pred,
                                               const float* __restrict__ gt,
                                               uint32_t* __restrict__ ws) {
  __shared__ float sPbuf[2][TILE];
  __shared__ float sGbuf[2][TILE];
  __shared__ uint32_t hP[2048];
  __shared__ uint32_t hG[2048];
  __shared__ uint32_t sCnt, sMin, sMax;
  const int row = blockIdx.x / BPR;
  const int blk = blockIdx.x % BPR;
  const float* gPb = pred + (size_t)row * N_COLS + (size_t)blk * EPB;
  const float* gGb = gt   + (size_t)row * N_COLS + (size_t)blk * EPB;
  for (int i = threadIdx.x; i < 2048; i += TPB) { hP[i] = 0; hG[i] = 0; }
  if (threadIdx.x == 0) { sCnt = 0; sMin = 0xFFFFFFFFu; sMax = 0; }
  __syncthreads();
  uint32_t cnt = 0, mn = 0xFFFFFFFFu, mx = 0;
  STREAM_TILES(
    for (int i = threadIdx.x; i < TILE; i += TPB) {
      float g = sG[i];
      if (g > 0.0f) {
        uint32_t kg = fkey(g);
        uint32_t kp = fkey(sP[i]);
        atomicAdd(&hP[kp >> 21], 1u);
        atomicAdd(&hG[kg >> 21], 1u);
        cnt++;
        mn = mn < kg ? mn : kg;
        mx = mx > kg ? mx : kg;
      }
    }
  )
  atomicAdd(&sCnt, cnt);
  atomicMin(&sMin, mn);
  atomicMax(&sMax, mx);
  __syncthreads();
  for (int i = threadIdx.x; i < 2048; i += TPB) {
    uint32_t v = hP[i]; if (v) atomicAdd(&ws[OFF_HISTP + row * 2048 + i], v);
    v = hG[i];          if (v) atomicAdd(&ws[OFF_HISTG + row * 2048 + i], v);
  }
  if (threadIdx.x == 0) {
    atomicAdd(&ws[OFF_CNT + row], sCnt);
    atomicMin(&ws[OFF_GMINK + row], sMin);
    atomicMax(&ws[OFF_GMAXK + row], sMax);
  }
}

__global__ void k_scan0(uint32_t* ws) {
  const int row = blockIdx.x;
  uint32_t* histP = ws + OFF_HISTP + (size_t)row * 2048;
  uint32_t* histG = ws + OFF_HISTG + (size_t)row * 2048;
  if (threadIdx.x == 0) {
    uint32_t M = ws[OFF_CNT + row];
    uint32_t rP = (M > 0) ? ((M - 1) >> 1) : 0;
    uint32_t rG = (M > 0) ? ((M - 1) - ((M - 1) >> 1)) : 0;
    uint32_t c = 0, b;
    for (b = 0; b < 2048; ++b) { uint32_t h = histP[b]; if (c + h > rP) { rP -= c; break; } c += h; }
    if (b >= 2048) b = 2047;
    ws[OFF_PREFP + row] = b;  ws[OFF_RANKP + row] = rP;
    c = 0;
    uint32_t bg;
    for (bg = 0; bg < 2048; ++bg) { uint32_t h = histG[bg]; if (c + h > rG) { rG -= c; break; } c += h; }
    if (bg >= 2048) bg = 2047;
    ws[OFF_PREFG + row] = bg; ws[OFF_RANKG + row] = rG;
  }
  __syncthreads();
  for (int i = threadIdx.x; i < 2048; i += TPB) { histP[i] = 0; histG[i] = 0; }
}

// Pass 2: level-1 (bits 20:10) histograms restricted to selected 11-bit prefix.
__global__ void __launch_bounds__(TPB) k_hist1(const float* __restrict__ pred,
                                               const float* __restrict__ gt,
                                               uint32_t* __restrict__ ws) {
  __shared__ float sPbuf[2][TILE];
  __shared__ float sGbuf[2][TILE];
  __shared__ uint32_t hP[2048];
  __shared__ uint32_t hG[2048];
  const int row = blockIdx.x / BPR;
  const int blk = blockIdx.x % BPR;
  const float* gPb = pred + (size_t)row * N_COLS + (size_t)blk * EPB;
  const float* gGb = gt   + (size_t)row * N_COLS + (size_t)blk * EPB;
  const uint32_t prefP = ws[OFF_PREFP + row];
  const uint32_t prefG = ws[OFF_PREFG + row];
  for (int i = threadIdx.x; i < 2048; i += TPB) { hP[i] = 0; hG[i] = 0; }
  __syncthreads();
  STREAM_TILES(
    for (int i = threadIdx.x; i < TILE; i += TPB) {
      float g = sG[i];
      if (g > 0.0f) {
        uint32_t kp = fkey(sP[i]);
        if ((kp >> 21) == prefP) atomicAdd(&hP[(kp >> 10) & 0x7FFu], 1u);
        uint32_t kg = fkey(g);
        if ((kg >> 21) == prefG) atomicAdd(&hG[(kg >> 10) & 0x7FFu], 1u);
      }
    }
  )
  for (int i = threadIdx.x; i < 2048; i += TPB) {
    uint32_t v = hP[i]; if (v) atomicAdd(&ws[OFF_HISTP + row * 2048 + i], v);
    v = hG[i];          if (v) atomicAdd(&ws[OFF_HISTG + row * 2048 + i], v);
  }
}

__global__ void k_scan1(uint32_t* ws) {
  const int row = blockIdx.x;
  uint32_t* histP = ws + OFF_HISTP + (size_t)row * 2048;
  uint32_t* histG = ws + OFF_HISTG + (size_t)row * 2048;
  if (threadIdx.x == 0) {
    uint32_t rP = ws[OFF_RANKP + row];
    uint32_t c = 0, b;
    for (b = 0; b < 2048; ++b) { uint32_t h = histP[b]; if (c + h > rP) { rP -= c; break; } c += h; }
    if (b >= 2048) b = 2047;
    ws[OFF_PREFP + row] = (ws[OFF_PREFP + row] << 11) | b;
    ws[OFF_RANKP + row] = rP;
    uint32_t rG = ws[OFF_RANKG + row];
    c = 0;
    uint32_t bg;
    for (bg = 0; bg < 2048; ++bg) { uint32_t h = histG[bg]; if (c + h > rG) { rG -= c; break; } c += h; }
    if (bg >= 2048) bg = 2047;
    ws[OFF_PREFG + row] = (ws[OFF_PREFG + row] << 11) | bg;
    ws[OFF_RANKG + row] = rG;
  }
  __syncthreads();
  for (int i = threadIdx.x; i < 2048; i += TPB) { histP[i] = 0; histG[i] = 0; }
}

// Pass 3: level-2 (bits 9:0) histograms restricted to selected 22-bit prefix.
__global__ void __launch_bounds__(TPB) k_hist2(const float* __restrict__ pred,
                                               const float* __restrict__ gt,
                                               uint32_t* __restrict__ ws) {
  __shared__ float sPbuf[2][TILE];
  __shared__ float sGbuf[2][TILE];
  __shared__ uint32_t hP[1024];
  __shared__ uint32_t hG[1024];
  const int row = blockIdx.x / BPR;
  const int blk = blockIdx.x % BPR;
  const float* gPb = pred + (size_t)row * N_COLS + (size_t)blk * EPB;
  const float* gGb = gt   + (size_t)row * N_COLS + (size_t)blk * EPB;
  const uint32_t prefP = ws[OFF_PREFP + row];
  const uint32_t prefG = ws[OFF_PREFG + row];
  for (int i = threadIdx.x; i < 1024; i += TPB) { hP[i] = 0; hG[i] = 0; }
  __syncthreads();
  STREAM_TILES(
    for (int i = threadIdx.x; i < TILE; i += TPB) {
      float g = sG[i];
      if (g > 0.0f) {
        uint32_t kp = fkey(sP[i]);
        if ((kp >> 10) == prefP) atomicAdd(&hP[kp & 0x3FFu], 1u);
        uint32_t kg = fkey(g);
        if ((kg >> 10) == prefG) atomicAdd(&hG[kg & 0x3FFu], 1u);
      }
    }
  )
  for (int i = threadIdx.x; i < 1024; i += TPB) {
    uint32_t v = hP[i]; if (v) atomicAdd(&ws[OFF_HISTP + row * 2048 + i], v);
    v = hG[i];          if (v) atomicAdd(&ws[OFF_HISTG + row * 2048 + i], v);
  }
}

__global__ void k_scan2(uint32_t* ws) {
  const int row = blockIdx.x;
  const uint32_t* histP = ws + OFF_HISTP + (size_t)row * 2048;
  const uint32_t* histG = ws + OFF_HISTG + (size_t)row * 2048;
  if (threadIdx.x == 0) {
    float* wf = (float*)ws;
    uint32_t M = ws[OFF_CNT + row];
    if (M == 0) {
      wf[OFF_TP + row] = 0.0f;
      wf[OFF_TG + row] = 0.0f;
      return;
    }
    uint32_t rP = ws[OFF_RANKP + row];
    uint32_t c = 0, b;
    for (b = 0; b < 1024; ++b) { uint32_t h = histP[b]; if (c + h > rP) break; c += h; }
    if (b >= 1024) b = 1023;
    wf[OFF_TP + row] = funkey((ws[OFF_PREFP + row] << 10) | b);   // median(pred | valid)
    uint32_t rG = ws[OFF_RANKG + row];
    c = 0;
    uint32_t bg;
    for (bg = 0; bg < 1024; ++bg) { uint32_t h = histG[bg]; if (c + h > rG) break; c += h; }
    if (bg >= 1024) bg = 1023;
    float gmed = funkey((ws[OFF_PREFG + row] << 10) | bg);
    wf[OFF_TG + row] = 1.0f / gmed;                               // median(gi | valid)
  }
}

// Pass 4: mean-abs-deviation sums for pred and gi (per-block deterministic slots).
__global__ void __launch_bounds__(TPB) k_sums(const float* __restrict__ pred,
                                              const float* __restrict__ gt,
                                              uint32_t* __restrict__ ws) {
  __shared__ float sPbuf[2][TILE];
  __shared__ float sGbuf[2][TILE];
  __shared__ float sRed[TPB];
  const int row = blockIdx.x / BPR;
  const int blk = blockIdx.x % BPR;
  const float* gPb = pred + (size_t)row * N_COLS + (size_t)blk * EPB;
  const float* gGb = gt   + (size_t)row * N_COLS + (size_t)blk * EPB;
  const float* wf = (const float*)ws;
  const float tP = wf[OFF_TP + row];
  const float tG = wf[OFF_TG + row];
  float ap = 0.0f, ag = 0.0f;
  STREAM_TILES(
    for (int i = threadIdx.x; i < TILE; i += TPB) {
      float g = sG[i];
      if (g > 0.0f) {
        ap += fabsf(sP[i] - tP);
        ag += fabsf(1.0f / g - tG);
      }
    }
  )
  float rp = block_reduce_sum(ap, sRed);
  float rg = block_reduce_sum(ag, sRed);
  if (threadIdx.x == 0) {
    ((float*)ws)[OFF_PSP + blockIdx.x] = rp;
    ((float*)ws)[OFF_PSG + blockIdx.x] = rg;
  }
}

// Pass 5: err = |pa - ga| * mask, summed per block.
__global__ void __launch_bounds__(TPB) k_err(const float* __restrict__ pred,
                                             const float* __restrict__ gt,
                                             uint32_t* __restrict__ ws) {
  __shared__ float sPbuf[2][TILE];
  __shared__ float sGbuf[2][TILE];
  __shared__ float sRed[TPB];
  const int row = blockIdx.x / BPR;
  const int blk = blockIdx.x % BPR;
  const float* gPb = pred + (size_t)row * N_COLS + (size_t)blk * EPB;
  const float* gGb = gt   + (size_t)row * N_COLS + (size_t)blk * EPB;
  const float* wf = (const float*)ws;
  const uint32_t M = ws[OFF_CNT + row];
  const float m = (M > 0) ? (float)M : 1.0f;
  const float tP = wf[OFF_TP + row];
  const float tG = wf[OFF_TG + row];
  float sump = 0.0f, sumg = 0.0f;
  for (int j = 0; j < BPR; ++j) {        // deterministic scalar reduce of block slots
    sump += wf[OFF_PSP + row * BPR + j];
    sumg += wf[OFF_PSG + row * BPR + j];
  }
  const float sp = sump / m;
  const float ssp = (sp > 0.0f) ? sp : 1.0f;          // safe_s for pred
  const float gimin = 1.0f / funkey(ws[OFF_GMAXK + row]);
  const float gimax = 1.0f / funkey(ws[OFF_GMINK + row]);
  const float range = gimax - gimin;
  const float t_gn = (tG - gimin) / range;
  const float s_gn = (sumg / m) / range;
  const float ssg = (s_gn > 0.0f) ? s_gn : 1.0f;      // safe_s for gn

  float acc = 0.0f;
  STREAM_TILES(
    for (int i = threadIdx.x; i < TILE; i += TPB) {
      float g = sG[i];
      if (g > 0.0f) {
        float pa = (sP[i] - tP) / ssp;
        float gn = (1.0f / g - gimin) / range;
        float ga = (gn - t_gn) / ssg;
        acc += fabsf(pa - ga);
      }
    }
  )
  float r = block_reduce_sum(acc, sRed);
  if (threadIdx.x == 0) ((float*)ws)[OFF_PERR + blockIdx.x] = r;
}

__global__ void k_final(const uint32_t* __restrict__ ws, float* __restrict__ out) {
  if (threadIdx.x == 0 && blockIdx.x == 0) {
    const float* wf = (const float*)ws;
    float total = 0.0f;
    int bv = 0;
    for (int r = 0; r < B_ROWS; ++r) {
      uint32_t M = ws[OFF_CNT + r];
      float e = 0.0f;
      for (int j = 0; j < BPR; ++j) e += wf[OFF_PERR + r * BPR + j];
      if (M > 0) { total += e / (2.0f * (float)M); bv++; }
    }
    out[0] = (bv > 0) ? (total / (float)bv) : 0.0f;
  }
}

extern "C" void kernel_launch(void* const* d_in, const int* in_sizes, int n_in,
                              void* d_out, int out_size, void* d_ws, size_t ws_size,
                              hipStream_t stream) {
  (void)in_sizes; (void)n_in; (void)out_size; (void)ws_size;
  const float* pred = (const float*)d_in[0];
  const float* gt   = (const float*)d_in[1];
  uint32_t* ws = (uint32_t*)d_ws;
  float* out = (float*)d_out;

  k_init <<<(WS_DWORDS + TPB - 1) / TPB, TPB, 0, stream>>>(ws);
  k_hist0<<<NBLK,   TPB, 0, stream>>>(pred, gt, ws);
  k_scan0<<<B_ROWS, TPB, 0, stream>>>(ws);
  k_hist1<<<NBLK,   TPB, 0, stream>>>(pred, gt, ws);
  k_scan1<<<B_ROWS, TPB, 0, stream>>>(ws);
  k_hist2<<<NBLK,   TPB, 0, stream>>>(pred, gt, ws);
  k_scan2<<<B_ROWS, TPB, 0, stream>>>(ws);
  k_sums <<<NBLK,   TPB, 0, stream>>>(pred, gt, ws);
  k_err  <<<NBLK,   TPB, 0, stream>>>(pred, gt, ws);
  k_final<<<1, 64, 0, stream>>>(ws, out);
}